// PerturbedSelect_12524124635964
// MI455X (gfx1250) — compile-verified
//
#include <hip/hip_runtime.h>
#include <stdint.h>

typedef __attribute__((ext_vector_type(2))) float v2f;
typedef __attribute__((ext_vector_type(8))) float v8f;

#define NUM_SAMPLES 1000
#define K_OUT 64
#define D_DIM 4096
#define BATCH 2048
#define N_HALF 131072000u   // (1000*64*4096)/2
#define SIGMA  9.9998f      // max(0.1, 10.0*0.99998)
#define SPLIT  16           // waves per GEMM block (intra-block split-K)
#define KCHUNK (D_DIM / SPLIT)

// ---------------------------------------------------------------------------
// Threefry-2x32, 20 rounds, matching JAX's implementation exactly.
// ---------------------------------------------------------------------------
__device__ __forceinline__ uint32_t rotl32(uint32_t x, int n) {
    return (x << n) | (x >> (32 - n));
}

__device__ __forceinline__ void threefry2x32(uint32_t x0, uint32_t x1,
                                             uint32_t& o0, uint32_t& o1) {
    const uint32_t ks0 = 0u;          // key(42) -> (0, 42)
    const uint32_t ks1 = 42u;
    const uint32_t ks2 = ks0 ^ ks1 ^ 0x1BD11BDAu;
    x0 += ks0; x1 += ks1;
#define TF_QR(r) { x0 += x1; x1 = rotl32(x1, r); x1 ^= x0; }
    TF_QR(13) TF_QR(15) TF_QR(26) TF_QR(6)
    x0 += ks1; x1 += ks2 + 1u;
    TF_QR(17) TF_QR(29) TF_QR(16) TF_QR(24)
    x0 += ks2; x1 += ks0 + 2u;
    TF_QR(13) TF_QR(15) TF_QR(26) TF_QR(6)
    x0 += ks0; x1 += ks1 + 3u;
    TF_QR(17) TF_QR(29) TF_QR(16) TF_QR(24)
    x0 += ks1; x1 += ks2 + 4u;
    TF_QR(13) TF_QR(15) TF_QR(26) TF_QR(6)
    x0 += ks2; x1 += ks0 + 5u;
#undef TF_QR
    o0 = x0; o1 = x1;
}

// JAX bits->uniform in [tiny, 1), then w = -log(u) (Exp(1) variate).
__device__ __forceinline__ float bits_to_nexplog(uint32_t bits) {
    const float TINY = 1.1754943508222875e-38f;  // finfo(f32).tiny
    float u = __uint_as_float((bits >> 9) | 0x3f800000u) - 1.0f; // [0,1)
    u = u * (1.0f - TINY) + TINY;                                // [tiny,1)
    u = fmaxf(u, TINY);
    return -__logf(u);
}

// ---------------------------------------------------------------------------
// Kernel 0a: zero the selections accumulator.
// Kernel 0b: cexp[k,d] = exp(-logits[k,d]/sigma)  (exponential-race weights)
// ---------------------------------------------------------------------------
__global__ void zero_sel_kernel(float* __restrict__ sel, int n) {
    int i = blockIdx.x * blockDim.x + threadIdx.x;
    if (i < n) sel[i] = 0.0f;
}

__global__ void prep_cexp_kernel(const float* __restrict__ logits,
                                 float* __restrict__ cexp, int n) {
    int i = blockIdx.x * blockDim.x + threadIdx.x;
    if (i < n) cexp[i] = __expf(-logits[i] * (1.0f / SIGMA));
}

// ---------------------------------------------------------------------------
// Kernel 1: one block per (n, k) with n in [0,500): each threefry call yields
// the exponential variates for samples n and n+500 at the same (k, d).
//   argmax_d( l + sigma*gumbel ) == argmin_d( Exp(1) * exp(-l/sigma) )
// Two simultaneous argmin reductions over d, then two atomic adds of 1/1000
// (order-independent: every add to a given address is the same constant).
// ---------------------------------------------------------------------------
__global__ __launch_bounds__(256)
void gumbel_argmax_kernel(const float* __restrict__ cexp,
                          float* __restrict__ sel) {
    const int k = blockIdx.x & (K_OUT - 1);
    const int n = blockIdx.x >> 6;                  // 0..499
    const int tid = threadIdx.x;

    const uint32_t base = ((uint32_t)n * K_OUT + (uint32_t)k) * D_DIM;
    const float* crow = cexp + (size_t)k * D_DIM;

    float best0 = INFINITY, best1 = INFINITY;
    int   idx0  = 0,        idx1  = 0;

    for (int d = tid; d < D_DIM; d += 256) {
        uint32_t o0, o1;
        const uint32_t i = base + (uint32_t)d;      // i < N_HALF by construction
        threefry2x32(i, i + N_HALF, o0, o1);
        const float c  = crow[d];
        const float s0 = bits_to_nexplog(o0) * c;   // sample n
        const float s1 = bits_to_nexplog(o1) * c;   // sample n+500
        if (s0 < best0) { best0 = s0; idx0 = d; }   // d increases: first-min
        if (s1 < best1) { best1 = s1; idx1 = d; }
    }

    // wave32 reduction (tie-break: prefer lower index)
    #pragma unroll
    for (int off = 16; off > 0; off >>= 1) {
        float ov0 = __shfl_down(best0, off, 32);
        int   oi0 = __shfl_down(idx0,  off, 32);
        float ov1 = __shfl_down(best1, off, 32);
        int   oi1 = __shfl_down(idx1,  off, 32);
        if (ov0 < best0 || (ov0 == best0 && oi0 < idx0)) { best0 = ov0; idx0 = oi0; }
        if (ov1 < best1 || (ov1 == best1 && oi1 < idx1)) { best1 = ov1; idx1 = oi1; }
    }

    __shared__ float sb0[8]; __shared__ int si0[8];
    __shared__ float sb1[8]; __shared__ int si1[8];
    const int wid = tid >> 5;
    if ((tid & 31) == 0) { sb0[wid] = best0; si0[wid] = idx0;
                           sb1[wid] = best1; si1[wid] = idx1; }
    __syncthreads();

    if (tid == 0) {
        float b0 = sb0[0], b1 = sb1[0];
        int   j0 = si0[0], j1 = si1[0];
        #pragma unroll
        for (int w = 1; w < 8; ++w) {
            if (sb0[w] < b0 || (sb0[w] == b0 && si0[w] < j0)) { b0 = sb0[w]; j0 = si0[w]; }
            if (sb1[w] < b1 || (sb1[w] == b1 && si1[w] < j1)) { b1 = sb1[w]; j1 = si1[w]; }
        }
        const float inc = 1.0f / (float)NUM_SAMPLES;
        atomicAdd(&sel[(size_t)k * D_DIM + j0], inc);
        atomicAdd(&sel[(size_t)k * D_DIM + j1], inc);
    }
}

// ---------------------------------------------------------------------------
// Kernel 2: Y[2048,64] = X[2048,4096] @ sel[64,4096]^T via f32 WMMA.
// 128 blocks (one 16-row M strip each) x 16 waves; wave w owns K chunk
// [w*256, w*256+256). Partial accumulators are combined through LDS in a
// fixed w=0..15 order (deterministic), then stored once.
// ISA f32 layouts (16x16x4):
//   A 16x4 : lane<16 -> M=lane, v[0]=K0, v[1]=K1 ; lane>=16 -> K2,K3
//   B 4x16 : lane<16 -> N=lane, v[0]=K0, v[1]=K1 ; lane>=16 -> K2,K3
//   C 16x16: vgpr r, lane<16 -> (M=r, N=lane) ; lane>=16 -> (M=r+8, N=lane-16)
// ---------------------------------------------------------------------------
__global__ __launch_bounds__(32 * SPLIT)
void wmma_gemm_kernel(const float* __restrict__ X,
                      const float* __restrict__ sel,
                      float* __restrict__ Y) {
    __shared__ float red[SPLIT * 1024];     // [w][nb][r][lane], 64 KB

    const int lane  = threadIdx.x & 31;
    const int wid   = threadIdx.x >> 5;     // 0..15 : K-split index
    const int row   = lane & 15;
    const int khalf = lane >> 4;            // K offset {0,1} vs {2,3}
    const int m0    = blockIdx.x * 16;
    const int k0    = wid * KCHUNK;

    v8f acc0 = {}, acc1 = {}, acc2 = {}, acc3 = {};

    const float* xrow = X   + (size_t)(m0 + row) * D_DIM + k0 + khalf * 2;
    const float* bp   = sel + (size_t)row        * D_DIM + k0 + khalf * 2;

    #pragma unroll 8
    for (int k = 0; k < KCHUNK; k += 4) {
        if ((k & 31) == 0) __builtin_prefetch(xrow + k + 128, 0, 1); // global_prefetch_b8
        v2f a  = *(const v2f*)(xrow + k);
        v2f b0 = *(const v2f*)(bp + 0 * 16 * D_DIM + k);
        v2f b1 = *(const v2f*)(bp + 1 * 16 * D_DIM + k);
        v2f b2 = *(const v2f*)(bp + 2 * 16 * D_DIM + k);
        v2f b3 = *(const v2f*)(bp + 3 * 16 * D_DIM + k);
        acc0 = __builtin_amdgcn_wmma_f32_16x16x4_f32(false, a, false, b0, (short)0, acc0, false, false);
        acc1 = __builtin_amdgcn_wmma_f32_16x16x4_f32(false, a, false, b1, (short)0, acc1, false, false);
        acc2 = __builtin_amdgcn_wmma_f32_16x16x4_f32(false, a, false, b2, (short)0, acc2, false, false);
        acc3 = __builtin_amdgcn_wmma_f32_16x16x4_f32(false, a, false, b3, (short)0, acc3, false, false);
    }

    // park this wave's partial tiles in LDS
    float* dst = red + wid * 1024 + lane;
    #pragma unroll
    for (int r = 0; r < 8; ++r) {
        dst[0 * 256 + r * 32] = acc0[r];
        dst[1 * 256 + r * 32] = acc1[r];
        dst[2 * 256 + r * 32] = acc2[r];
        dst[3 * 256 + r * 32] = acc3[r];
    }
    __syncthreads();

    // deterministic reduction: 1024 output positions, fixed w-order sum
    for (int p = threadIdx.x; p < 1024; p += 32 * SPLIT) {
        float s = 0.0f;
        #pragma unroll
        for (int w = 0; w < SPLIT; ++w) s += red[w * 1024 + p];
        const int nb = p >> 8;
        const int r  = (p >> 5) & 7;
        const int l  = p & 31;
        const int rr = m0 + (l >> 4) * 8 + r;
        const int cc = nb * 16 + (l & 15);
        Y[(size_t)rr * K_OUT + cc] = s;
    }
}

// ---------------------------------------------------------------------------
extern "C" void kernel_launch(void* const* d_in, const int* in_sizes, int n_in,
                              void* d_out, int out_size, void* d_ws, size_t ws_size,
                              hipStream_t stream) {
    const float* X      = (const float*)d_in[0];   // [2048, 4096]
    const float* logits = (const float*)d_in[1];   // [64, 4096]
    float* Y = (float*)d_out;                      // [2048, 64]

    const int sel_elems = K_OUT * D_DIM;           // 262144
    float* sel  = (float*)d_ws;                    // [64,4096] accumulator (1 MB)
    float* cexp = sel + sel_elems;                 // [64,4096] race weights (1 MB)

    zero_sel_kernel<<<(sel_elems + 255) / 256, 256, 0, stream>>>(sel, sel_elems);
    prep_cexp_kernel<<<(sel_elems + 255) / 256, 256, 0, stream>>>(logits, cexp, sel_elems);

    // 500 sample-pairs x 64 rows = 32000 blocks
    gumbel_argmax_kernel<<<(NUM_SAMPLES / 2) * K_OUT, 256, 0, stream>>>(cexp, sel);

    // 2048/16 = 128 M-strips, 16 split-K waves each
    wmma_gemm_kernel<<<BATCH / 16, 32 * SPLIT, 0, stream>>>(X, sel, Y);
}